// MNB_3470333575853
// MI455X (gfx1250) — compile-verified
//
#include <hip/hip_runtime.h>

// Problem constants (match reference.py)
constexpr int L      = 50;     // SEQ_LEN
constexpr int P      = 4096;   // N_PHRASES
constexpr int BLOCK  = 128;    // 4 wave32s per workgroup
constexpr int WAVES_PER_BLOCK = BLOCK / 32;
constexpr int TOKPAD = 64;     // padded token slots per phrase (50 -> 64)

#if defined(__gfx1250__)
typedef __attribute__((address_space(1))) int gas_int;   // global AS int
typedef __attribute__((address_space(3))) int las_int;   // LDS AS int
#endif

// ---------------------------------------------------------------------------
// CDNA5 async global->LDS copy of one dword, per-lane addressing.
// Tracked by ASYNCcnt; no VGPR return path. Builtin if available, else the
// raw gfx1250 mnemonic (GV mode: vdst = LDS byte offset, vaddr = 64-bit VA).
// ---------------------------------------------------------------------------
__device__ __forceinline__ void async_copy_b32_to_lds(const int* gsrc, int* ldst) {
#if defined(__gfx1250__) && __has_builtin(__builtin_amdgcn_global_load_async_to_lds_b32)
  __builtin_amdgcn_global_load_async_to_lds_b32(
      (gas_int*)gsrc,        // addrspace(1) int*, const cast away (C-style)
      (las_int*)ldst,        // addrspace(3) int*
      /*imm offset*/ 0, /*cpol*/ 0);
#elif defined(__gfx1250__)
  unsigned lds_off = (unsigned)(size_t)(las_int*)ldst;
  unsigned long long ga = (unsigned long long)(size_t)gsrc;
  asm volatile("global_load_async_to_lds_b32 %0, %1, off"
               :: "v"(lds_off), "v"(ga)
               : "memory");
#else
  *ldst = *gsrc;  // host-parse / non-gfx1250 fallback (never codegen'd on host)
#endif
}

__device__ __forceinline__ void wait_all_async() {
#if defined(__gfx1250__) && __has_builtin(__builtin_amdgcn_s_wait_asynccnt)
  __builtin_amdgcn_s_wait_asynccnt(0);
  asm volatile("" ::: "memory");   // compiler fence so LDS reads don't hoist
#elif defined(__gfx1250__)
  asm volatile("s_wait_asynccnt 0" ::: "memory");
#endif
}

// ---------------------------------------------------------------------------
// One wave32 per phrase. Stage the phrase's 50 token ids (low dword of each
// int64) into LDS via async copies, dedup (first-occurrence = binary
// presence), gather-sum W, wave-reduce, add bias.
// ---------------------------------------------------------------------------
__global__ void __launch_bounds__(BLOCK) presence_linear_kernel(
    const int*   __restrict__ text32,  // [L][P] int64 viewed as int32 pairs
    const float* __restrict__ W,       // [VOCAB]
    const float* __restrict__ bias,    // [1]
    float*       __restrict__ out)     // [P]
{
  __shared__ int toks[WAVES_PER_BLOCK * TOKPAD];

  const int lane = threadIdx.x & 31;
  const int wave = threadIdx.x >> 5;
  const int p    = blockIdx.x * WAVES_PER_BLOCK + wave;
  int* base = &toks[wave * TOKPAD];

  // Stage tokens: lane handles positions (lane) and (lane+32).
  // text is int64 little-endian; low dword of element (t*P + p) is at
  // int32 index 2*(t*P + p). Values < 50000 fit in the low dword.
  async_copy_b32_to_lds(text32 + 2 * (lane * P + p), base + lane);
  if (lane + 32 < L)
    async_copy_b32_to_lds(text32 + 2 * ((lane + 32) * P + p), base + lane + 32);
  wait_all_async();   // only this wave's own staged data is read -> no barrier

  // Token at position t contributes W[id] once iff no earlier position j<t
  // holds the same id (Counter-keys / binary presence semantics).
  const int  my0 = base[lane];
  const int  t1  = lane + 32;
  const int  my1 = (t1 < L) ? base[t1] : -1;
  bool dup0 = false, dup1 = false;
#pragma unroll
  for (int j = 0; j < L; ++j) {
    const int tj = base[j];           // uniform LDS broadcast read
    dup0 = dup0 || ((j < lane) && (tj == my0));
    dup1 = dup1 || ((j < t1)   && (tj == my1));
  }

  float acc = 0.0f;
  if (!dup0)              acc += W[my0];   // random gather, W is L2-resident
  if (my1 >= 0 && !dup1)  acc += W[my1];

  // wave32 butterfly reduction
#pragma unroll
  for (int m = 16; m >= 1; m >>= 1)
    acc += __shfl_xor(acc, m, 32);

  if (lane == 0)
    out[p] = acc + bias[0];
}

// ---------------------------------------------------------------------------
extern "C" void kernel_launch(void* const* d_in, const int* in_sizes, int n_in,
                              void* d_out, int out_size, void* d_ws, size_t ws_size,
                              hipStream_t stream) {
  (void)in_sizes; (void)n_in; (void)out_size; (void)d_ws; (void)ws_size;

  const int*   text32 = (const int*)d_in[0];    // int64 [L,P] viewed as int32 pairs
  const float* W      = (const float*)d_in[1];  // float32 [1,V]
  const float* bias   = (const float*)d_in[2];  // float32 [1]
  float*       out    = (float*)d_out;          // float32 [P,1]

  const dim3 grid(P / WAVES_PER_BLOCK);   // 1024 workgroups
  const dim3 block(BLOCK);                // 4 wave32s each
  presence_linear_kernel<<<grid, block, 0, stream>>>(text32, W, bias, out);
}